// Collisions_69681549410972
// MI455X (gfx1250) — compile-verified
//
#include <hip/hip_runtime.h>

// f(x,y,vx,vy): NX=128, NY=64, NVX=64, NVY=64, fp32.
// One workgroup (64 threads = 2 wave32) per (x,y) cell; the whole 64x64
// velocity tile lives in LDS. Single pass over HBM: 128MB read + 128MB write
// (~11us at 23.3 TB/s). Moments computed with V_WMMA_F32_16X16X4_F32.
#define NX  128
#define NY  64
#define NVX 64
#define NVY 64
#define PAD 65   // 64+1: bank-conflict-free for both sweep directions

typedef float v2f __attribute__((ext_vector_type(2)));
typedef float v8f __attribute__((ext_vector_type(8)));

__launch_bounds__(64)
__global__ void collisions_fused_kernel(const float* __restrict__ nu_fp_p,
                                        const float* __restrict__ nu_K,
                                        const float* __restrict__ f_in,
                                        const float* __restrict__ dt_p,
                                        const float* __restrict__ vx_g,
                                        const float* __restrict__ vy_g,
                                        const float* __restrict__ f_mx,
                                        const float* __restrict__ dvx_p,
                                        const float* __restrict__ dvy_p,
                                        float* __restrict__ f_out)
{
    __shared__ float tile[NVX * PAD];   // tile[vx][vy], padded
    __shared__ float cpb [NVX * PAD];   // Thomas cp scratch, same layout
    __shared__ float Msh[3][64];        // per-row moments S0,S1,S2
    __shared__ float vshx[64];
    __shared__ float vshy[64];
    __shared__ float nprof_sh;

    const int t    = threadIdx.x;       // 0..63
    const int lane = t & 31;            // wave32 lane
    const int wave = t >> 5;            // 0 or 1
    const int b    = blockIdx.x;        // x*NY + y
    const long long base = (long long)b * (NVX * NVY);

    const float nu_fp = nu_fp_p[0];
    const float dt    = dt_p[0];
    const float dvx   = dvx_p[0];
    const float dvy   = dvy_p[0];
    const float nu_dt = nu_fp * dt;

    vshx[t] = vx_g[t];
    vshy[t] = vy_g[t];

    // ---- Phase 1: load 64x64 tile, coalesced float4 ----
    {
        const float4* src = (const float4*)(f_in + base);
        #pragma unroll
        for (int i = 0; i < 16; ++i) {
            int flat4 = i * 64 + t;           // 0..1023
            float4 v  = src[flat4];
            int flat  = flat4 * 4;
            int ivx   = flat >> 6;
            int ivy   = flat & 63;
            tile[ivx * PAD + ivy + 0] = v.x;
            tile[ivx * PAD + ivy + 1] = v.y;
            tile[ivx * PAD + ivy + 2] = v.z;
            tile[ivx * PAD + ivy + 3] = v.w;
        }
    }
    __syncthreads();

    const int m    = lane & 15;   // A-matrix row within 16-group / B column
    const int half = lane >> 4;
    const int k0   = 2 * half;    // K sub-index held by this half-wave

    // Branchless B-weight selectors, hoisted out of all loops.
    // B column n: n=0 -> 1, n=1 -> v, n=2 -> v^2, n>=3 -> 0
    // bw(v) = w0 + v*(w1 + w2*v)
    const float w0 = (m == 0) ? 1.0f : 0.0f;
    const float w1 = (m == 1) ? 1.0f : 0.0f;
    const float w2 = (m == 2) ? 1.0f : 0.0f;

    const int row0 = wave * 32 + m;        // group 2*wave
    const int row1 = wave * 32 + 16 + m;   // group 2*wave+1

    // ================= direction 1: along vx (rows = vy) =================
    // D(16x16) += A(16x4) * B(4x16); one shared B fragment per chunk,
    // two accumulators (two 16-row groups) per wave.
    {
        v8f acc0 = {}, acc1 = {};
        #pragma unroll
        for (int c4 = 0; c4 < 16; ++c4) {
            int kb = c4 * 4 + k0;          // vx (K) base for this half-wave
            float v0 = vshx[kb + 0];
            float v1 = vshx[kb + 1];
            v2f bw;
            bw[0] = w0 + v0 * (w1 + w2 * v0);
            bw[1] = w0 + v1 * (w1 + w2 * v1);
            v2f a0, a1;
            a0[0] = tile[(kb + 0) * PAD + row0];
            a0[1] = tile[(kb + 1) * PAD + row0];
            a1[0] = tile[(kb + 0) * PAD + row1];
            a1[1] = tile[(kb + 1) * PAD + row1];
            acc0 = __builtin_amdgcn_wmma_f32_16x16x4_f32(
                     false, a0, false, bw, (short)0, acc0, false, false);
            acc1 = __builtin_amdgcn_wmma_f32_16x16x4_f32(
                     false, a1, false, bw, (short)0, acc1, false, false);
        }
        if (m < 3) {
            #pragma unroll
            for (int r = 0; r < 8; ++r) {
                Msh[m][wave * 32 +      half * 8 + r] = acc0[r];
                Msh[m][wave * 32 + 16 + half * 8 + r] = acc1[r];
            }
        }
    }
    __syncthreads();

    // Thomas solve along vx; thread t owns the row vy = t.
    {
        float S0 = Msh[0][t], S1 = Msh[1][t], S2 = Msh[2][t];
        float vbar = S1 * dvx;
        float v0t  = (S2 - 2.0f * vbar * S1 + vbar * vbar * S0) * dvx;
        float idv2 = nu_dt * v0t / (dvx * dvx);
        float h    = nu_dt / (2.0f * dvx);
        float bb   = 1.0f + 2.0f * idv2;

        float cp_prev, dp_prev;
        {
            float vp = vshx[1];
            float c0 = -idv2 - (vp - vbar) * h;
            float rb = __builtin_amdgcn_rcpf(bb);
            cp_prev  = c0 * rb;
            dp_prev  = tile[0 * PAD + t] * rb;
            cpb[0 * PAD + t]  = cp_prev;
            tile[0 * PAD + t] = dp_prev;
        }
        for (int k = 1; k < 64; ++k) {
            float vm = vshx[k - 1];
            float vp = (k == 63) ? vshx[0] : vshx[k + 1];
            float ak = -idv2 + (vm - vbar) * h;
            float ck = -idv2 - (vp - vbar) * h;
            float denom = bb - ak * cp_prev;
            float rd  = __builtin_amdgcn_rcpf(denom);
            float cpk = ck * rd;
            float dpk = (tile[k * PAD + t] - ak * dp_prev) * rd;
            cpb[k * PAD + t]  = cpk;
            tile[k * PAD + t] = dpk;
            cp_prev = cpk; dp_prev = dpk;
        }
        float xn = dp_prev;   // x[63] already stored
        for (int k = 62; k >= 0; --k) {
            float xk = tile[k * PAD + t] - cpb[k * PAD + t] * xn;
            tile[k * PAD + t] = xk;
            xn = xk;
        }
    }
    __syncthreads();

    // ================= direction 2: along vy (rows = vx) =================
    {
        v8f acc0 = {}, acc1 = {};
        #pragma unroll
        for (int c4 = 0; c4 < 16; ++c4) {
            int kb = c4 * 4 + k0;          // vy (K) base
            float v0 = vshy[kb + 0];
            float v1 = vshy[kb + 1];
            v2f bw;
            bw[0] = w0 + v0 * (w1 + w2 * v0);
            bw[1] = w0 + v1 * (w1 + w2 * v1);
            v2f a0, a1;
            a0[0] = tile[row0 * PAD + kb + 0];
            a0[1] = tile[row0 * PAD + kb + 1];
            a1[0] = tile[row1 * PAD + kb + 0];
            a1[1] = tile[row1 * PAD + kb + 1];
            acc0 = __builtin_amdgcn_wmma_f32_16x16x4_f32(
                     false, a0, false, bw, (short)0, acc0, false, false);
            acc1 = __builtin_amdgcn_wmma_f32_16x16x4_f32(
                     false, a1, false, bw, (short)0, acc1, false, false);
        }
        if (m < 3) {
            #pragma unroll
            for (int r = 0; r < 8; ++r) {
                Msh[m][wave * 32 +      half * 8 + r] = acc0[r];
                Msh[m][wave * 32 + 16 + half * 8 + r] = acc1[r];
            }
        }
    }
    __syncthreads();

    // Thomas solve along vy; thread t owns the row vx = t.
    {
        float S0 = Msh[0][t], S1 = Msh[1][t], S2 = Msh[2][t];
        float vbar = S1 * dvy;
        float v0t  = (S2 - 2.0f * vbar * S1 + vbar * vbar * S0) * dvy;
        float idv2 = nu_dt * v0t / (dvy * dvy);
        float h    = nu_dt / (2.0f * dvy);
        float bb   = 1.0f + 2.0f * idv2;

        float cp_prev, dp_prev;
        {
            float vp = vshy[1];
            float c0 = -idv2 - (vp - vbar) * h;
            float rb = __builtin_amdgcn_rcpf(bb);
            cp_prev  = c0 * rb;
            dp_prev  = tile[t * PAD + 0] * rb;
            cpb[t * PAD + 0]  = cp_prev;
            tile[t * PAD + 0] = dp_prev;
        }
        for (int k = 1; k < 64; ++k) {
            float vm = vshy[k - 1];
            float vp = (k == 63) ? vshy[0] : vshy[k + 1];
            float ak = -idv2 + (vm - vbar) * h;
            float ck = -idv2 - (vp - vbar) * h;
            float denom = bb - ak * cp_prev;
            float rd  = __builtin_amdgcn_rcpf(denom);
            float cpk = ck * rd;
            float dpk = (tile[t * PAD + k] - ak * dp_prev) * rd;
            cpb[t * PAD + k]  = cpk;
            tile[t * PAD + k] = dpk;
            cp_prev = cpk; dp_prev = dpk;
        }
        float xn = dp_prev;
        for (int k = 62; k >= 0; --k) {
            float xk = tile[t * PAD + k] - cpb[t * PAD + k] * xn;
            tile[t * PAD + k] = xk;
            xn = xk;
        }
    }
    __syncthreads();

    // ---- density n_prof = sum(tile) * dvx * dvy ----
    {
        float s = 0.0f;
        for (int k = 0; k < 64; ++k) s += tile[t * PAD + k];
        Msh[0][t] = s;
    }
    __syncthreads();
    if (t == 0) {
        float s = 0.0f;
        for (int k = 0; k < 64; ++k) s += Msh[0][k];
        nprof_sh = s;
    }
    __syncthreads();

    // ---- BGK blend and coalesced float4 store ----
    {
        float e     = __expf(-dt * nu_K[b]);
        float blend = nprof_sh * dvx * dvy * (1.0f - e);
        const float4* fm4 = (const float4*)f_mx;
        float4* dst = (float4*)(f_out + base);
        #pragma unroll
        for (int i = 0; i < 16; ++i) {
            int flat4 = i * 64 + t;
            int flat  = flat4 * 4;
            int ivx   = flat >> 6;
            int ivy   = flat & 63;
            float4 fm = fm4[flat4];
            float4 o;
            o.x = tile[ivx * PAD + ivy + 0] * e + fm.x * blend;
            o.y = tile[ivx * PAD + ivy + 1] * e + fm.y * blend;
            o.z = tile[ivx * PAD + ivy + 2] * e + fm.z * blend;
            o.w = tile[ivx * PAD + ivy + 3] * e + fm.w * blend;
            dst[flat4] = o;
        }
    }
}

extern "C" void kernel_launch(void* const* d_in, const int* in_sizes, int n_in,
                              void* d_out, int out_size, void* d_ws, size_t ws_size,
                              hipStream_t stream) {
    // setup_inputs order:
    // 0: nu_fp (1)   1: nu_K (NX*NY)   2: f (NX*NY*NVX*NVY)   3: dt (1)
    // 4: vx (64)     5: vy (64)        6: f_mx (64*64)        7: dvx (1)  8: dvy (1)
    const float* nu_fp = (const float*)d_in[0];
    const float* nu_K  = (const float*)d_in[1];
    const float* f     = (const float*)d_in[2];
    const float* dt    = (const float*)d_in[3];
    const float* vx    = (const float*)d_in[4];
    const float* vy    = (const float*)d_in[5];
    const float* f_mx  = (const float*)d_in[6];
    const float* dvx   = (const float*)d_in[7];
    const float* dvy   = (const float*)d_in[8];
    float* out = (float*)d_out;

    dim3 grid(NX * NY);   // 8192 tiles
    dim3 block(64);       // 2 wave32 per tile
    collisions_fused_kernel<<<grid, block, 0, stream>>>(
        nu_fp, nu_K, f, dt, vx, vy, f_mx, dvx, dvy, out);
}